// FMLinearBundle4_24086176596731
// MI455X (gfx1250) — compile-verified
//
#include <hip/hip_runtime.h>
#include <cstddef>

// ---------------------------------------------------------------------------
// GatedDeltaNet decode step (4 stacked layers), MI455X / gfx1250.
// Bandwidth-bound (~550 MB of f32 weights per call => ~24 us @ 23.3 TB/s).
// Big GEMVs use V_WMMA_F32_16X16X4_F32 (A = transposed W tile, B = x in col 0)
// so weight loads stay coalesced while feeding the CDNA5 matrix pipe.
// The x chunk is staged into LDS via GLOBAL_LOAD_ASYNC_TO_LDS (ASYNCcnt).
// ---------------------------------------------------------------------------

typedef __attribute__((ext_vector_type(2))) float v2f;
typedef __attribute__((ext_vector_type(8))) float v8f;

#define H_DIM    2048
#define NK       16
#define NV       32
#define DK       128
#define DV       128
#define KEY_DIM  (NK * DK)          // 2048
#define VAL_DIM  (NV * DV)          // 4096
#define QKVZ_N   (2 * KEY_DIM + 2 * VAL_DIM)   // 12288
#define CONV_DIM (2 * KEY_DIM + VAL_DIM)       // 8192
#define BA_N     (2 * NV)           // 64
#define EPSV     1e-6f

// workspace layout (floats)
#define O_X     0                           // x vector (2048)
#define O_P1    2048                        // qkvz+ba partials 4 x 12352
#define O_QKVZ  (O_P1 + 4 * 12352)          // reduced qkvz|ba (12352)
#define O_CONV  (O_QKVZ + 12352)            // conv output q|k|v (8192)
#define O_QH    (O_CONV + 8192)             // normalized q, 16x128
#define O_KH    (O_QH + 2048)               // normalized k, 16x128
#define O_SC    (O_KH + 2048)               // beta[32], decay[32]
#define O_O     (O_SC + 64)                 // o, 32x128
#define O_OG    (O_O + 4096)                // gated o, 4096
#define O_P2    (O_OG + 4096)               // out-proj partials 4 x 2048
// total: 92544 floats = 370 KB of d_ws

#if defined(__HIP_DEVICE_COMPILE__) && __has_builtin(__builtin_amdgcn_global_load_async_to_lds_b32)
#define HAVE_ASYNC_LDS 1
#else
#define HAVE_ASYNC_LDS 0
#endif

// ---------------------------------------------------------------------------
// GEMV y[c] = sum_r x[r] * W[r*ld + c] via WMMA f32 16x16x4.
// One wave owns a 16-column strip. Per iteration (4 rows):
//   A[m,k] = W[(r0+k)*ld + c0+m]  (transposed tile; VGPR loads are two
//            contiguous 64B row segments across the two half-waves)
//   B[k,0] = x[r0+k], other columns zero
//   D[m,0] += partial y for column c0+m
// A lane layout (32b A 16x4): VGPR0 lanes0-15=K0, lanes16-31=K2; VGPR1 K1/K3.
// D col 0 lives in lane 0 (M=0..7 -> acc[0..7]) and lane 16 (M=8..15).
// Grid: x = nstrips/4 (4 waves/block), y = row splits. Deterministic partials.
// x chunk (rowsPerSplit <= 1024 floats) staged in LDS once per block.
// ---------------------------------------------------------------------------
__global__ __launch_bounds__(128)
void gemv_wmma_kernel(const float* __restrict__ x, const float* __restrict__ W,
                      int ld, int rowsPerSplit,
                      float* __restrict__ pbuf, int pstride, int pcol0)
{
    __shared__ float xs[1024];

    const int lane  = threadIdx.x & 31;
    const int wave  = threadIdx.x >> 5;
    const int strip = blockIdx.x * 4 + wave;       // grid sized exactly
    const int c0    = strip * 16;
    const int half  = lane >> 4;                   // 0: rows r,r+1  1: rows r+2,r+3
    const int col   = c0 + (lane & 15);
    const int r0    = blockIdx.y * rowsPerSplit;
    const int iters = rowsPerSplit >> 2;
    const bool bnz  = ((lane & 15) == 0);          // only N=0 column of B is live

    // --- stage x[r0 .. r0+rowsPerSplit) into LDS (async DMA path on CDNA5) ---
    {
        const float* gsrc = x + r0;
#if HAVE_ASYNC_LDS
        for (int idx = threadIdx.x; idx < rowsPerSplit; idx += 128) {
            __builtin_amdgcn_global_load_async_to_lds_b32(
                (__attribute__((address_space(1))) int*)(gsrc + idx),
                (__attribute__((address_space(3))) int*)(xs + idx),
                0, 0);
        }
        asm volatile("s_wait_asynccnt 0x0" ::: "memory");
#else
        for (int idx = threadIdx.x; idx < rowsPerSplit; idx += 128)
            xs[idx] = gsrc[idx];
#endif
        __syncthreads();
    }

    const float* wptr = W + (size_t)(r0 + 2 * half) * ld + col;

    v8f acc = {};
    for (int it = 0; it < iters; ++it) {
        v2f a, b;
        a.x = wptr[0];
        a.y = wptr[ld];
        // prefetch the weight stream ~16 rows ahead (speculative, safe at EOB)
        __builtin_prefetch(wptr + (size_t)16 * ld, 0, 0);
        wptr += (size_t)4 * ld;

        float x0 = xs[it * 4 + 2 * half];
        float x1 = xs[it * 4 + 2 * half + 1];
        b.x = bnz ? x0 : 0.0f;
        b.y = bnz ? x1 : 0.0f;

        acc = __builtin_amdgcn_wmma_f32_16x16x4_f32(
            false, a, false, b, (short)0, acc, false, false);
    }

    const int outbase = blockIdx.y * pstride + pcol0 + c0;
    if (lane == 0) {
        #pragma unroll
        for (int j = 0; j < 8; ++j) pbuf[outbase + j] = acc[j];
    }
    if (lane == 16) {
        #pragma unroll
        for (int j = 0; j < 8; ++j) pbuf[outbase + 8 + j] = acc[j];
    }
}

// ---------------------------------------------------------------------------
// Fused: reduce qkvz/ba partials, K=4 conv + SiLU, write new conv state,
// q/k L2 normalization, per-head scalars beta / decay. One block, 512 thr.
// ---------------------------------------------------------------------------
__global__ __launch_bounds__(512)
void mid_kernel(const float* __restrict__ pbuf, const float* __restrict__ Cin,
                const float* __restrict__ cw, const float* __restrict__ dtb,
                const float* __restrict__ Alog,
                float* __restrict__ qkvzr, float* __restrict__ convout,
                float* __restrict__ qhat, float* __restrict__ khat,
                float* __restrict__ scal, float* __restrict__ nC_out)
{
    const int t = threadIdx.x;

    // 1) reduce the 4 row-split partials (qkvz cols 0..12287, ba 12288..12351)
    for (int c = t; c < 12352; c += 512)
        qkvzr[c] = pbuf[c] + pbuf[12352 + c] + pbuf[2 * 12352 + c] + pbuf[3 * 12352 + c];
    __syncthreads();

    // 2) depthwise conv (window = [C0, C1, C2, mixed]) + SiLU; new conv state
    for (int c = t; c < CONV_DIM; c += 512) {
        float w0 = Cin[c];
        float w1 = Cin[CONV_DIM + c];
        float w2 = Cin[2 * CONV_DIM + c];
        float w3 = qkvzr[c];                      // mixed = qkv pre-conv
        const float* cwc = cw + (size_t)c * 4;
        float s = w0 * cwc[0] + w1 * cwc[1] + w2 * cwc[2] + w3 * cwc[3];
        s = s / (1.0f + __expf(-s));              // SiLU
        convout[c] = s;
        nC_out[c] = w1;
        nC_out[CONV_DIM + c] = w2;
        nC_out[2 * CONV_DIM + c] = w3;
    }
    __syncthreads();

    // 3) q/k L2 norm: 32 groups (16 q-heads then 16 k-heads) x 16 threads
    __shared__ float red[512];
    __shared__ float scale[32];
    const int g = t >> 4, j = t & 15;
    const float* src = convout + ((g < 16) ? g * DK : KEY_DIM + (g - 16) * DK);
    float ss = 0.0f;
    #pragma unroll
    for (int e = 0; e < 8; ++e) { float v = src[j * 8 + e]; ss += v * v; }
    red[t] = ss;
    __syncthreads();
    if (j == 0) {
        float s2 = 0.0f;
        #pragma unroll
        for (int e = 0; e < 16; ++e) s2 += red[(g << 4) + e];
        scale[g] = rsqrtf(s2 + EPSV);
    }
    __syncthreads();
    {
        float sc = scale[g];
        float* dst = (g < 16) ? (qhat + g * DK) : (khat + (g - 16) * DK);
        #pragma unroll
        for (int e = 0; e < 8; ++e) dst[j * 8 + e] = src[j * 8 + e] * sc;
    }

    // 4) per-head scalars: beta = sigmoid(b); decay = exp(-exp(Alog)*softplus(a+dtb))
    if (t < NV) {
        float b = qkvzr[QKVZ_N + t];
        float a = qkvzr[QKVZ_N + NV + t];
        float beta = 1.0f / (1.0f + __expf(-b));
        float xab = a + dtb[t];
        float sp = (xab > 20.0f) ? xab : log1pf(__expf(xab));
        scal[t] = beta;
        scal[NV + t] = __expf(-__expf(Alog[t]) * sp);
    }
}

// ---------------------------------------------------------------------------
// Delta-rule state update. Grid (32 heads, 4 row-groups) x 256 threads.
// Each wave owns 4 value-rows; lanes hold 4 contiguous DK elements (float4),
// row dot-products via wave32 shfl_xor butterfly.
// ---------------------------------------------------------------------------
__global__ __launch_bounds__(256)
void state_kernel(const float* __restrict__ Sin, const float* __restrict__ convout,
                  const float* __restrict__ qhat, const float* __restrict__ khat,
                  const float* __restrict__ scal,
                  float* __restrict__ Sout, float* __restrict__ obuf)
{
    const int h    = blockIdx.x;            // value head
    const int bg   = blockIdx.y;            // row group (32 rows each)
    const int wave = threadIdx.x >> 5;
    const int lane = threadIdx.x & 31;
    const int d    = lane * 4;

    const float* kh = khat + (h >> 1) * DK;
    const float* qh = qhat + (h >> 1) * DK;
    const float beta  = scal[h];
    const float decay = scal[NV + h];
    const float4 k4 = *(const float4*)(kh + d);
    const float4 q4 = *(const float4*)(qh + d);
    const float* vvec = convout + KEY_DIM + h * DV;

    #pragma unroll
    for (int i = 0; i < 4; ++i) {
        const int vd = bg * 32 + wave * 4 + i;
        const size_t off = (size_t)h * DV * DK + (size_t)vd * DK + d;
        float4 s4 = *(const float4*)(Sin + off);
        s4.x *= decay; s4.y *= decay; s4.z *= decay; s4.w *= decay;

        float kv = s4.x * k4.x + s4.y * k4.y + s4.z * k4.z + s4.w * k4.w;
        #pragma unroll
        for (int m = 16; m >= 1; m >>= 1) kv += __shfl_xor(kv, m, 32);

        const float delta = (vvec[vd] - kv) * beta;
        s4.x += delta * k4.x; s4.y += delta * k4.y;
        s4.z += delta * k4.z; s4.w += delta * k4.w;
        *(float4*)(Sout + off) = s4;

        float ov = s4.x * q4.x + s4.y * q4.y + s4.z * q4.z + s4.w * q4.w;
        #pragma unroll
        for (int m = 16; m >= 1; m >>= 1) ov += __shfl_xor(ov, m, 32);
        if (lane == 0) obuf[h * DV + vd] = ov * 0.08838834764831845f;  // DK^-0.5
    }
}

// ---------------------------------------------------------------------------
// Per-head RMS norm of o, * norm_w, * SiLU(z). Grid 32 x 128.
// ---------------------------------------------------------------------------
__global__ __launch_bounds__(128)
void ogate_kernel(const float* __restrict__ obuf, const float* __restrict__ qkvzr,
                  const float* __restrict__ nw, float* __restrict__ og)
{
    const int h = blockIdx.x, d = threadIdx.x;
    __shared__ float red[128];
    float v = obuf[h * DV + d];
    red[d] = v * v;
    __syncthreads();
    for (int s = 64; s > 0; s >>= 1) {
        if (d < s) red[d] += red[d + s];
        __syncthreads();
    }
    float sc = rsqrtf(red[0] * (1.0f / DV) + EPSV);
    float z = qkvzr[2 * KEY_DIM + h * DV + d];
    float silu = z / (1.0f + __expf(-z));
    og[h * DV + d] = v * sc * nw[d] * silu;
}

// ---------------------------------------------------------------------------
// Deterministic reduction of 4 row-split partials into the layer output.
// ---------------------------------------------------------------------------
__global__ __launch_bounds__(256)
void reduce4_kernel(const float* __restrict__ pbuf, float* __restrict__ dst,
                    int n, int stride)
{
    const int c = blockIdx.x * blockDim.x + threadIdx.x;
    if (c < n)
        dst[c] = pbuf[c] + pbuf[stride + c] + pbuf[2 * stride + c] + pbuf[3 * stride + c];
}

// ---------------------------------------------------------------------------
extern "C" void kernel_launch(void* const* d_in, const int* in_sizes, int n_in,
                              void* d_out, int out_size, void* d_ws, size_t ws_size,
                              hipStream_t stream)
{
    (void)in_sizes; (void)n_in; (void)out_size; (void)ws_size;

    const float* hs  = (const float*)d_in[0];
    const float* gs[4] = { (const float*)d_in[1], (const float*)d_in[3],
                           (const float*)d_in[5], (const float*)d_in[7] };
    const float* cs[4] = { (const float*)d_in[2], (const float*)d_in[4],
                           (const float*)d_in[6], (const float*)d_in[8] };
    const float* Wq  = (const float*)d_in[9];
    const float* Wba = (const float*)d_in[10];
    const float* cw  = (const float*)d_in[11];
    const float* dtb = (const float*)d_in[12];
    const float* Al  = (const float*)d_in[13];
    const float* nw  = (const float*)d_in[14];
    const float* Wo  = (const float*)d_in[15];

    float* out = (float*)d_out;
    float* ws  = (float*)d_ws;

    // seed x with hidden_states
    (void)hipMemcpyAsync(ws + O_X, hs, H_DIM * sizeof(float),
                         hipMemcpyDeviceToDevice, stream);

    for (int l = 0; l < 4; ++l) {
        float* nS = out + H_DIM + (size_t)l * (NV * DV * DK + 3 * CONV_DIM);
        float* nC = nS + NV * DV * DK;

        // qkvz GEMV: 768 strips x 4 row-splits (512 rows each)
        gemv_wmma_kernel<<<dim3(192, 4), 128, 0, stream>>>(
            ws + O_X, Wq + (size_t)l * H_DIM * QKVZ_N, QKVZ_N, 512,
            ws + O_P1, 12352, 0);
        // ba GEMV: 4 strips x 4 row-splits
        gemv_wmma_kernel<<<dim3(1, 4), 128, 0, stream>>>(
            ws + O_X, Wba + (size_t)l * H_DIM * BA_N, BA_N, 512,
            ws + O_P1, 12352, QKVZ_N);

        mid_kernel<<<1, 512, 0, stream>>>(
            ws + O_P1, cs[l], cw + (size_t)l * CONV_DIM * 4,
            dtb + l * NV, Al + l * NV,
            ws + O_QKVZ, ws + O_CONV, ws + O_QH, ws + O_KH, ws + O_SC, nC);

        state_kernel<<<dim3(32, 4), 256, 0, stream>>>(
            gs[l], ws + O_CONV, ws + O_QH, ws + O_KH, ws + O_SC, nS, ws + O_O);

        ogate_kernel<<<32, 128, 0, stream>>>(
            ws + O_O, ws + O_QKVZ, nw + l * DV, ws + O_OG);

        // output projection GEMV: 128 strips x 4 row-splits (1024 rows each)
        gemv_wmma_kernel<<<dim3(32, 4), 128, 0, stream>>>(
            ws + O_OG, Wo + (size_t)l * VAL_DIM * H_DIM, H_DIM, 1024,
            ws + O_P2, H_DIM, 0);

        float* ydst = (l == 3) ? out : (ws + O_X);
        reduce4_kernel<<<8, 256, 0, stream>>>(ws + O_P2, ydst, H_DIM, H_DIM);
    }
}